// ClosestPool1D_63969242906681
// MI455X (gfx1250) — compile-verified
//
#include <hip/hip_runtime.h>

typedef float v2f __attribute__((ext_vector_type(2)));
typedef float v8f __attribute__((ext_vector_type(8)));

constexpr int Npts   = 32768;
constexpr int Mpts   = 8192;
constexpr int C      = 256;
constexpr int NTILES = Npts / 16;   // 2048
constexpr int WAVES  = 8;           // 256 threads / wave32

// Merge two sorted top-2 pairs (d1<=d2, e1<=e2) -> smallest two, tie-break on index.
// All selects single-condition; bitwise bool ops (no short-circuit branches).
__device__ __forceinline__ void merge2(float& d1, int& i1, float& d2, int& i2,
                                       float e1, int j1, float e2, int j2) {
  bool ta = (d1 < e1) | ((d1 == e1) & (i1 <= j1));
  float w1 = ta ? d1 : e1;
  int   x1 = ta ? i1 : j1;
  float ra = ta ? d2 : e2;   // runner-up of winner
  int   ya = ta ? i2 : j2;
  float rb = ta ? e1 : d1;   // best of loser
  int   yb = ta ? j1 : i1;
  bool tb = (ra < rb) | ((ra == rb) & (ya <= yb));
  d1 = w1; i1 = x1;
  d2 = tb ? ra : rb;
  i2 = tb ? ya : yb;
}

__global__ __launch_bounds__(256) void closest_pool2(
    const float* __restrict__ srcF,  const float* __restrict__ tgtF,
    const float* __restrict__ srcCo, const float* __restrict__ tgtCo,
    const float* __restrict__ srcSc, const float* __restrict__ tgtSc,
    float* __restrict__ out)
{
  const int side = blockIdx.y;
  const float* feats   = side ? tgtF  : srcF;
  const float* coords  = side ? tgtCo : srcCo;
  const float* scoords = side ? tgtSc : srcSc;
  float* outp = out + (size_t)side * (size_t)Mpts * C;

  const int  tid  = threadIdx.x;
  // readfirstlane: compiler-visibly wave-uniform -> scalar loop control,
  // EXEC stays all-ones at the WMMA (ISA requirement).
  const int  wave = __builtin_amdgcn_readfirstlane(tid >> 5);
  const int  lane = tid & 31;
  const int  cs   = lane & 15;          // column slot (A-row / B-col select)
  const bool hiH  = (lane & 16) != 0;   // lane-half: selects K-pair for A/B
  const int  m0   = blockIdx.x << 4;    // 16 query rows per block

  // ---- A matrix 16x4 f32: row m = (-2x, -2y, -2z, 1). Loop-invariant. ----
  const float* sp = scoords + 3 * (m0 + cs);
  float sx = sp[0], sy = sp[1], sz = sp[2];
  v2f A;
  A.x = hiH ? (-2.0f * sz) : (-2.0f * sx);
  A.y = hiH ? 1.0f         : (-2.0f * sy);

  // Top-2 state: 8 rows per lane (rows v / v+8 by lane-half), column slot = cs.
  float d1[8], d2[8];
  int   i1[8], i2[8];
  #pragma unroll
  for (int v = 0; v < 8; ++v) { d1[v] = 3.0e38f; d2[v] = 3.0e38f; i1[v] = 0; i2[v] = 0; }

  #pragma unroll 2
  for (int t = wave; t < NTILES; t += WAVES) {
    const int nn = (t << 4) + cs;               // this lane's candidate column index
    const float* cp = coords + 3 * nn;
    float x = cp[0], y = cp[1], z = cp[2];
    __builtin_prefetch(cp + 3 * 16 * WAVES, 0, 3);   // next tile for this wave
    float w = fmaf(z, z, fmaf(y, y, x * x));    // |b|^2
    // B matrix 4x16 f32 (mirrors A): lanes 0-15: K0,K1 ; lanes 16-31: K2,K3
    v2f B;
    B.x = hiH ? z : x;
    B.y = hiH ? w : y;
    v8f acc = {};
    // D[v] @ lane l: rows v (lanes 0-15) / v+8 (lanes 16-31), col = l%16
    v8f D = __builtin_amdgcn_wmma_f32_16x16x4_f32(
        /*neg_a=*/false, A, /*neg_b=*/false, B,
        /*c_mod=*/(short)0, acc, /*reuse_a=*/false, /*reuse_b=*/false);
    #pragma unroll
    for (int v = 0; v < 8; ++v) {
      float d  = D[v];
      bool  l1 = d < d1[v];
      bool  l2 = d < d2[v];
      // branch-proof update, reusing the compares (no fminf canonicalize):
      //   new d1 = l1 ? d : d1                      -> v_cndmask (reuses v_cmp)
      //   new d2 = second-smallest of {d, d1, d2}   -> v_med3_num_f32
      float nd1 = l1 ? d : d1[v];
      float nd2 = __builtin_amdgcn_fmed3f(d, d1[v], d2[v]);
      // index updates: flat single-condition selects
      i2[v] = l2 ? nn    : i2[v];   // tentative runner-up
      i2[v] = l1 ? i1[v] : i2[v];   // displaced best becomes runner-up
      i1[v] = l1 ? nn    : i1[v];
      d1[v] = nd1;
      d2[v] = nd2;
    }
  }

  // ---- butterfly-merge the 16 column slots (masks < 16 stay within lane-half) ----
  #pragma unroll
  for (int mask = 1; mask <= 8; mask <<= 1) {
    #pragma unroll
    for (int v = 0; v < 8; ++v) {
      float e1 = __shfl_xor(d1[v], mask, 32);
      float e2 = __shfl_xor(d2[v], mask, 32);
      int   j1 = __shfl_xor(i1[v], mask, 32);
      int   j2 = __shfl_xor(i2[v], mask, 32);
      merge2(d1[v], i1[v], d2[v], i2[v], e1, j1, e2, j2);
    }
  }

  // ---- combine the 8 waves' per-row partials through LDS ----
  __shared__ float sD1[WAVES][16], sD2[WAVES][16];
  __shared__ int   sI1[WAVES][16], sI2[WAVES][16];
  __shared__ int   sidx[16];

  if (cs == 0) {                       // lanes 0 and 16 hold the reduced results
    const int rbase = hiH ? 8 : 0;
    #pragma unroll
    for (int v = 0; v < 8; ++v) {
      sD1[wave][rbase + v] = d1[v];
      sD2[wave][rbase + v] = d2[v];
      sI1[wave][rbase + v] = i1[v];
      sI2[wave][rbase + v] = i2[v];
    }
  }
  __syncthreads();

  if (tid < 16) {
    float D1 = sD1[0][tid], D2 = sD2[0][tid];
    int   I1 = sI1[0][tid], I2 = sI2[0][tid];
    #pragma unroll
    for (int w2 = 1; w2 < WAVES; ++w2)
      merge2(D1, I1, D2, I2, sD1[w2][tid], sI1[w2][tid], sD2[w2][tid], sI2[w2][tid]);
    sidx[tid] = I2;                    // second-nearest index
  }
  __syncthreads();

  // ---- fused gather: out[m0+r, :] = feats[sidx[r], :], 256 f32 = 64 float4 ----
  const float4* f4 = reinterpret_cast<const float4*>(feats);
  float4*       o4 = reinterpret_cast<float4*>(outp);
  const int c4 = tid & 63;
  const int rq = tid >> 6;
  #pragma unroll
  for (int it = 0; it < 4; ++it) {
    int r = rq + (it << 2);
    int srow = sidx[r];
    o4[(size_t)(m0 + r) * (C / 4) + c4] = f4[(size_t)srow * (C / 4) + c4];
  }
}

extern "C" void kernel_launch(void* const* d_in, const int* in_sizes, int n_in,
                              void* d_out, int out_size, void* d_ws, size_t ws_size,
                              hipStream_t stream) {
  const float* src        = (const float*)d_in[0];
  const float* tgt        = (const float*)d_in[1];
  const float* src_coords = (const float*)d_in[2];
  const float* tgt_coords = (const float*)d_in[3];
  const float* src_sc     = (const float*)d_in[4];
  const float* tgt_sc     = (const float*)d_in[5];
  float* out = (float*)d_out;

  dim3 grid(Mpts / 16, 2);   // 512 row-tiles x 2 sides
  closest_pool2<<<grid, 256, 0, stream>>>(src, tgt, src_coords, tgt_coords,
                                          src_sc, tgt_sc, out);
  (void)in_sizes; (void)n_in; (void)out_size; (void)d_ws; (void)ws_size;
}